// GCN_layer_17145509446345
// MI455X (gfx1250) — compile-verified
//
#include <hip/hip_runtime.h>
#include <hip/hip_bf16.h>

typedef __attribute__((ext_vector_type(2))) float v2f;
typedef __attribute__((ext_vector_type(8))) float v8f;

// ---------------------------------------------------------------------------
// Edge preprocessing
// ---------------------------------------------------------------------------
__global__ void k_cvt_idx(const long long* __restrict__ e64,
                          int* __restrict__ e32, int n) {
    int i = blockIdx.x * blockDim.x + threadIdx.x;
    if (i < n) e32[i] = (int)e64[i];
}

__global__ void k_deg_init(float* __restrict__ deg, int n) {
    int i = blockIdx.x * blockDim.x + threadIdx.x;
    if (i < n) deg[i] = 1.0f;              // self-loop contributes 1 to every node
}

__global__ void k_deg_edges(const int* __restrict__ dst,
                            float* __restrict__ deg, int E) {
    int e = blockIdx.x * blockDim.x + threadIdx.x;
    if (e < E) unsafeAtomicAdd(&deg[dst[e]], 1.0f);
}

__global__ void k_dinv(float* __restrict__ deg, int n) {
    int i = blockIdx.x * blockDim.x + threadIdx.x;
    if (i < n) deg[i] = rsqrtf(deg[i]);    // deg >= 1 always (self loop)
}

__global__ void k_norm(const int* __restrict__ src, const int* __restrict__ dst,
                       const float* __restrict__ dinv,
                       float* __restrict__ norm, int E) {
    int e = blockIdx.x * blockDim.x + threadIdx.x;
    if (e < E) norm[e] = dinv[src[e]] * dinv[dst[e]];
}

// ---------------------------------------------------------------------------
// Dense GEMM: out[N,COUT] = (RELU? max(A,0):A)[N,CIN] @ W[CIN,COUT]
// One wave32 per 16x16 output tile, native fp32 WMMA 16x16x4, K fully unrolled.
// A layout (ISA 7.12.2, 32-bit A 16x4): lane%16 = row, (lane>>4)*2 = K base.
// B layout (4x16): lane%16 = col, same K base.
// D layout (16x16 f32, 8 VGPRs): vgpr v -> M = (lane>>4)*8 + v, N = lane%16.
// ---------------------------------------------------------------------------
template <int CIN, int COUT, bool RELU>
__global__ __launch_bounds__(32)
void k_gemm(const float* __restrict__ A, const float* __restrict__ W,
            float* __restrict__ out) {
    const int lane = threadIdx.x;
    const int l16  = lane & 15;
    const int hi   = lane >> 4;            // 0 or 1 -> K half
    const int row  = blockIdx.x * 16 + l16;
    const int col  = blockIdx.y * 16 + l16;

    const float* Arow = A + (size_t)row * CIN;
    v8f c = {0.f, 0.f, 0.f, 0.f, 0.f, 0.f, 0.f, 0.f};

#pragma unroll
    for (int k = 0; k < CIN; k += 4) {
        const int kb = k + hi * 2;                 // even -> 8B-aligned v2f load
        v2f a = *(const v2f*)(Arow + kb);
        if (RELU) { a[0] = fmaxf(a[0], 0.f); a[1] = fmaxf(a[1], 0.f); }
        v2f b;
        b[0] = W[kb * COUT + col];
        b[1] = W[(kb + 1) * COUT + col];
        // 8 args: (neg_a, A, neg_b, B, c_mod, C, reuse_a, reuse_b)
        c = __builtin_amdgcn_wmma_f32_16x16x4_f32(
                false, a, false, b, (short)0, c, false, false);
    }

#pragma unroll
    for (int v = 0; v < 8; ++v) {
        const int m = blockIdx.x * 16 + hi * 8 + v;
        out[(size_t)m * COUT + col] = c[v];
    }
}

// ---------------------------------------------------------------------------
// agg[i,c] = bias[c] + dinv[i]^2 * hw[i,c]   (self-loop term + bias)
// ---------------------------------------------------------------------------
__global__ void k_agg_init(const float* __restrict__ hw,
                           const float* __restrict__ dinv,
                           const float* __restrict__ bias,
                           float* __restrict__ agg, int n, int Cout) {
    int t = blockIdx.x * blockDim.x + threadIdx.x;
    if (t < n * Cout) {
        int i  = t / Cout;
        int ch = t - i * Cout;
        float di = dinv[i];
        agg[t] = bias[ch] + di * di * hw[t];
    }
}

// ---------------------------------------------------------------------------
// agg[dst[e], :] += norm[e] * hw[src[e], :]
// one thread per (edge, 4 channels): float4 gather + 4 fp32 atomics (L2-resident)
// ---------------------------------------------------------------------------
__global__ void k_agg_edges(const float* __restrict__ hw,
                            const int* __restrict__ src,
                            const int* __restrict__ dst,
                            const float* __restrict__ norm,
                            float* __restrict__ agg, int E, int Cout) {
    const int C4 = Cout >> 2;
    int t = blockIdx.x * blockDim.x + threadIdx.x;
    if (t >= E * C4) return;
    int e = t / C4;
    int q = t - e * C4;
    int s = src[e];
    int d = dst[e];
    float w = norm[e];
    const float4 v = *(const float4*)(hw + (size_t)s * Cout + q * 4);
    float* ap = agg + (size_t)d * Cout + q * 4;
    unsafeAtomicAdd(ap + 0, w * v.x);
    unsafeAtomicAdd(ap + 1, w * v.y);
    unsafeAtomicAdd(ap + 2, w * v.z);
    unsafeAtomicAdd(ap + 3, w * v.w);
}

// ---------------------------------------------------------------------------
extern "C" void kernel_launch(void* const* d_in, const int* in_sizes, int n_in,
                              void* d_out, int out_size, void* d_ws, size_t ws_size,
                              hipStream_t stream) {
    (void)n_in; (void)out_size; (void)ws_size;

    const float*     x   = (const float*)d_in[0];
    const long long* ei  = (const long long*)d_in[1];
    const float*     W1  = (const float*)d_in[2];
    const float*     b1  = (const float*)d_in[3];
    const float*     W2  = (const float*)d_in[4];
    const float*     b2  = (const float*)d_in[5];
    const float*     W3  = (const float*)d_in[6];
    const float*     b3  = (const float*)d_in[7];
    float*           out = (float*)d_out;

    const int N = in_sizes[0] / 12;     // 100000 (multiple of 16)
    const int E = in_sizes[1] / 2;      // 1600000

    char* ws = (char*)d_ws;
    float* dinv  = (float*)ws;  ws += sizeof(float) * (size_t)N;
    float* hw    = (float*)ws;  ws += sizeof(float) * (size_t)N * 128;
    float* agg1  = (float*)ws;  ws += sizeof(float) * (size_t)N * 64;
    float* agg2  = (float*)ws;  ws += sizeof(float) * (size_t)N * 128;
    int*   idx32 = (int*)ws;    ws += sizeof(int)   * (size_t)E * 2;
    float* norm  = (float*)ws;
    int* src32 = idx32;
    int* dst32 = idx32 + E;

    const int B = 256;

    // --- edge prep: int32 indices, degrees, symmetric norm ----------------
    k_cvt_idx  <<<(2 * E + B - 1) / B, B, 0, stream>>>(ei, idx32, 2 * E);
    k_deg_init <<<(N + B - 1) / B, B, 0, stream>>>(dinv, N);
    k_deg_edges<<<(E + B - 1) / B, B, 0, stream>>>(dst32, dinv, E);
    k_dinv     <<<(N + B - 1) / B, B, 0, stream>>>(dinv, N);
    k_norm     <<<(E + B - 1) / B, B, 0, stream>>>(src32, dst32, dinv, norm, E);

    // --- layer 1: 12 -> 64 -------------------------------------------------
    k_gemm<12, 64, false><<<dim3(N / 16, 64 / 16), 32, 0, stream>>>(x, W1, hw);
    k_agg_init <<<(N * 64 + B - 1) / B, B, 0, stream>>>(hw, dinv, b1, agg1, N, 64);
    k_agg_edges<<<(E * 16 + B - 1) / B, B, 0, stream>>>(hw, src32, dst32, norm, agg1, E, 64);

    // --- layer 2: 64 -> 128 (ReLU fused into A load) ----------------------
    k_gemm<64, 128, true><<<dim3(N / 16, 128 / 16), 32, 0, stream>>>(agg1, W2, hw);
    k_agg_init <<<(N * 128 + B - 1) / B, B, 0, stream>>>(hw, dinv, b2, agg2, N, 128);
    k_agg_edges<<<(E * 32 + B - 1) / B, B, 0, stream>>>(hw, src32, dst32, norm, agg2, E, 128);

    // --- layer 3: 128 -> 96 (ReLU fused), writes d_out --------------------
    k_gemm<128, 96, true><<<dim3(N / 16, 96 / 16), 32, 0, stream>>>(agg2, W3, hw);
    k_agg_init <<<(N * 96 + B - 1) / B, B, 0, stream>>>(hw, dinv, b3, out, N, 96);
    k_agg_edges<<<(E * 24 + B - 1) / B, B, 0, stream>>>(hw, src32, dst32, norm, out, E, 96);
}